// GraphSAGE_NET_19542101197276
// MI455X (gfx1250) — compile-verified
//
#include <hip/hip_runtime.h>

typedef __attribute__((ext_vector_type(2))) float v2f;
typedef __attribute__((ext_vector_type(8))) float v8f;

#define NODE_DIM 128   // K for both layers (D == H == 128)

// ---------------------------------------------------------------- utilities
__global__ void zero_f32(float* __restrict__ p, int n) {
    int i = blockIdx.x * blockDim.x + threadIdx.x;
    if (i < n) p[i] = 0.0f;
}

// one thread per edge: count in-degree of dst
__global__ void deg_count(const int* __restrict__ dst, float* __restrict__ deg, int E) {
    int e = blockIdx.x * blockDim.x + threadIdx.x;
    if (e < E) atomicAdd(deg + dst[e], 1.0f);
}

// deg[i] -> 1/max(deg[i],1)
__global__ void inv_deg(float* __restrict__ deg, int n) {
    int i = blockIdx.x * blockDim.x + threadIdx.x;
    if (i < n) deg[i] = 1.0f / fmaxf(deg[i], 1.0f);
}

// one wave per edge; lane l moves features [4l, 4l+4) (128 feats / 32 lanes)
__global__ void scatter_add(const float* __restrict__ feat,
                            const int* __restrict__ src,
                            const int* __restrict__ dst,
                            float* __restrict__ agg, int E) {
    int gt   = blockIdx.x * blockDim.x + threadIdx.x;
    int edge = gt >> 5;
    int lane = gt & 31;
    if (edge >= E) return;
    int s = src[edge];
    int d = dst[edge];
    const float4 v = *(const float4*)(feat + (size_t)s * NODE_DIM + lane * 4);
    float* p = agg + (size_t)d * NODE_DIM + lane * 4;
    atomicAdd(p + 0, v.x);
    atomicAdd(p + 1, v.y);
    atomicAdd(p + 2, v.z);
    atomicAdd(p + 3, v.w);
}

// ------------------------------------------------------- fused SAGE linear
// out[row, n] = act( (invdeg[row]*Aagg[row,:]) @ Wl[:, n]
//                  + Aself[row,:] @ Wr[:, n] + bias[n] )
// One wave -> one 16x16 output tile via V_WMMA_F32_16X16X4_F32.
// NCOLS/16 waves per block cover the full column range of a 16-row tile.
template <int NCOLS, int RELU>
__global__ __launch_bounds__(32 * (NCOLS / 16)) void sage_gemm(
    const float* __restrict__ Aagg,    // [N, 128]  (unscaled sums)
    const float* __restrict__ Aself,   // [N, 128]
    const float* __restrict__ invdeg,  // [N]
    const float* __restrict__ Wl,      // [128, NCOLS]
    const float* __restrict__ Wr,      // [128, NCOLS]
    const float* __restrict__ bias,    // [NCOLS]
    float* __restrict__ out,           // [N, NCOLS]
    int nrows) {
    const int lane  = threadIdx.x & 31;
    const int wave  = threadIdx.x >> 5;
    const int row0  = blockIdx.x * 16;
    const int lrow  = lane & 15;     // M (for A) / N (for B,C)
    const int half  = lane >> 4;     // 0: K={0,1}, 1: K={2,3}
    const int n     = wave * 16 + lrow;

    int arow = row0 + lrow;
    if (arow >= nrows) arow = nrows - 1;        // clamp reads; EXEC stays all-1
    const float scale = invdeg[arow];

    const float* __restrict__ aAgg  = Aagg  + (size_t)arow * NODE_DIM;
    const float* __restrict__ aSelf = Aself + (size_t)arow * NODE_DIM;

    v8f acc = {};

    // agg @ Wl  (mean normalization fused into the A fragment)
    #pragma unroll 4
    for (int k = 0; k < NODE_DIM; k += 4) {
        const int kk = k + half * 2;
        v2f a = *(const v2f*)(aAgg + kk);
        a *= scale;
        v2f b;
        b.x = Wl[(size_t)kk * NCOLS + n];
        b.y = Wl[(size_t)(kk + 1) * NCOLS + n];
        acc = __builtin_amdgcn_wmma_f32_16x16x4_f32(false, a, false, b,
                                                    (short)0, acc, false, false);
    }
    // self @ Wr
    #pragma unroll 4
    for (int k = 0; k < NODE_DIM; k += 4) {
        const int kk = k + half * 2;
        v2f a = *(const v2f*)(aSelf + kk);
        v2f b;
        b.x = Wr[(size_t)kk * NCOLS + n];
        b.y = Wr[(size_t)(kk + 1) * NCOLS + n];
        acc = __builtin_amdgcn_wmma_f32_16x16x4_f32(false, a, false, b,
                                                    (short)0, acc, false, false);
    }

    const float bv = bias[n];
    #pragma unroll
    for (int r = 0; r < 8; ++r) {
        const int m = r + half * 8;
        const int row = row0 + m;
        if (row < nrows) {
            float v = acc[r] + bv;
            if (RELU) v = fmaxf(v, 0.0f);
            out[(size_t)row * NCOLS + n] = v;
        }
    }
}

// ---------------------------------------------------------------- launcher
extern "C" void kernel_launch(void* const* d_in, const int* in_sizes, int n_in,
                              void* d_out, int out_size, void* d_ws, size_t ws_size,
                              hipStream_t stream) {
    const float* x   = (const float*)d_in[0];           // [N, 128]
    const int*   ei  = (const int*)d_in[1];             // [2, E]
    const float* W1l = (const float*)d_in[2];           // [128, 128]
    const float* W1r = (const float*)d_in[3];           // [128, 128]
    const float* b1  = (const float*)d_in[4];           // [128]
    const float* W2l = (const float*)d_in[5];           // [128, 64]
    const float* W2r = (const float*)d_in[6];           // [128, 64]
    const float* b2  = (const float*)d_in[7];           // [64]
    float* out = (float*)d_out;

    const int N = in_sizes[0] / NODE_DIM;               // 10000
    const int E = in_sizes[1] / 2;                      // 640000
    const int* src = ei;
    const int* dst = ei + E;

    float* agg = (float*)d_ws;                          // [N, 128]
    float* h   = agg + (size_t)N * NODE_DIM;            // [N, 128]
    float* deg = h   + (size_t)N * NODE_DIM;            // [N]

    const int zn     = N * NODE_DIM;
    const int tiles  = (N + 15) / 16;

    // ---- layer 1 -------------------------------------------------------
    zero_f32<<<(zn + 255) / 256, 256, 0, stream>>>(agg, zn);
    zero_f32<<<(N + 255) / 256, 256, 0, stream>>>(deg, N);
    deg_count<<<(E + 255) / 256, 256, 0, stream>>>(dst, deg, E);
    scatter_add<<<(E + 7) / 8, 256, 0, stream>>>(x, src, dst, agg, E);
    inv_deg<<<(N + 255) / 256, 256, 0, stream>>>(deg, N);
    sage_gemm<128, 1><<<tiles, 256, 0, stream>>>(agg, x, deg, W1l, W1r, b1, h, N);

    // ---- layer 2 (reuse agg buffer and inverted degrees) ---------------
    zero_f32<<<(zn + 255) / 256, 256, 0, stream>>>(agg, zn);
    scatter_add<<<(E + 7) / 8, 256, 0, stream>>>(h, src, dst, agg, E);
    sage_gemm<64, 0><<<tiles, 128, 0, stream>>>(agg, h, deg, W2l, W2r, b2, out, N);
}